// HierarchicalSoftmax_48687749267665
// MI455X (gfx1250) — compile-verified
//
#include <hip/hip_runtime.h>
#include <math.h>

#define NTOK 50257
#define NHID 1024
#define NTPC 225
#define NCLS 224
#define BATCH 1024

typedef float v2f __attribute__((ext_vector_type(2)));
typedef float v4f __attribute__((ext_vector_type(4)));
typedef float v8f __attribute__((ext_vector_type(8)));

// ---------------- bucketing ----------------

__global__ void hs_zero_counts(int* counts) {
  int t = threadIdx.x;
  if (t < NCLS) counts[t] = 0;
}

__global__ void hs_bucket(const int* __restrict__ labels, int* __restrict__ counts,
                          int* __restrict__ list) {
  int s = blockIdx.x * blockDim.x + threadIdx.x;
  if (s >= BATCH) return;
  int c = labels[s] / NTPC;            // top class, 0..223
  int pos = atomicAdd(&counts[c], 1);
  list[c * BATCH + pos] = s;
}

// ---------------- top GEMM: logits = inputs @ Wt + bt ----------------
// One wave per 16x16 output tile. WMMA f32 16x16x4, K = 1024.
// A frag (16x4 f32): lanes 0-15 hold K = kb+0,kb+1 ; lanes 16-31 hold K = kb+2,kb+3.
// B frag (4x16 f32): same K split across lane halves, N = lane&15.
// C/D (16x16 f32): VGPR r -> M = r (lanes 0-15) / r+8 (lanes 16-31), N = lane&15.
__global__ void __launch_bounds__(32) hs_top_gemm(const float* __restrict__ inputs,
                                                  const float* __restrict__ Wt,
                                                  const float* __restrict__ bt,
                                                  float* __restrict__ out) {
  const int NT = NCLS / 16;                 // 14 N tiles
  int mt = blockIdx.x / NT;
  int nt = blockIdx.x % NT;
  int lane   = threadIdx.x;
  int lane15 = lane & 15;
  int khalf  = (lane >> 4) * 2;             // 0 or 2
  int col    = nt * 16 + lane15;            // < 224 always
  const float* arow = inputs + (size_t)(mt * 16 + lane15) * NHID;

  v8f acc = {};
  for (int kb = 0; kb < NHID; kb += 4) {
    int k0 = kb + khalf;
    v2f a = *(const v2f*)(arow + k0);       // A[m][k0], A[m][k0+1]
    v2f b;
    b.x = Wt[(size_t)k0 * NCLS + col];      // B[k0][n]
    b.y = Wt[(size_t)(k0 + 1) * NCLS + col];
    acc = __builtin_amdgcn_wmma_f32_16x16x4_f32(false, a, false, b,
                                                (short)0, acc, false, false);
  }
  float bias = bt[col];
  int mhi = (lane >> 4) * 8;
#pragma unroll
  for (int r = 0; r < 8; ++r) {
    int m = mt * 16 + r + mhi;
    out[(size_t)m * NCLS + col] = acc[r] + bias;
  }
}

// ---------------- top softmax -> target-class prob ----------------

__global__ void __launch_bounds__(256) hs_top_softmax(const float* __restrict__ logits,
                                                      const int* __restrict__ labels,
                                                      float* __restrict__ top_prob) {
  __shared__ float sh[256];
  int s = blockIdx.x, t = threadIdx.x;
  float l = (t < NCLS) ? logits[(size_t)s * NCLS + t] : -INFINITY;
  sh[t] = l; __syncthreads();
  for (int k = 128; k > 0; k >>= 1) { if (t < k) sh[t] = fmaxf(sh[t], sh[t + k]); __syncthreads(); }
  float mx = sh[0]; __syncthreads();
  float e = (t < NCLS) ? expf(l - mx) : 0.f;
  sh[t] = e; __syncthreads();
  for (int k = 128; k > 0; k >>= 1) { if (t < k) sh[t] += sh[t + k]; __syncthreads(); }
  float sum = sh[0];
  if (t == 0) {
    int c = labels[s] / NTPC;
    top_prob[s] = expf(logits[(size_t)s * NCLS + c] - mx) / sum;
  }
}

// ---------------- bottom GEMM, grouped by class ----------------
// One block (8 waves) per class: gather <=16 sample rows into LDS,
// GEMM [16 x 1024] @ [1024 x 225] with WMMA f32 16x16x4, scatter rows out.
__global__ void __launch_bounds__(256) hs_bottom_gemm(const float* __restrict__ inputs,
                                                      const float* __restrict__ Wb,
                                                      const float* __restrict__ bb,
                                                      const int* __restrict__ counts,
                                                      const int* __restrict__ list,
                                                      float* __restrict__ blogits) {
  __shared__ float As[16 * NHID];           // 64 KB LDS A-tile
  int c = blockIdx.x;
  int cnt = counts[c];
  if (cnt == 0) return;                     // uniform across block
  const float* Wc = Wb + (size_t)c * NHID * NTPC;
  int t = threadIdx.x;
  int wave = t >> 5, lane = t & 31;
  int lane15 = lane & 15;
  int khalf = (lane >> 4) * 2;

  for (int base = 0; base < cnt; base += 16) {
    // Gather A tile: 16 threads per row, float4 strided copies, zero-pad.
    {
      int r  = t >> 4;                      // 0..15
      int j0 = t & 15;
      int sidx = (base + r < cnt) ? list[c * BATCH + base + r] : -1;
      const float* src = inputs + (size_t)(sidx < 0 ? 0 : sidx) * NHID;
      for (int j = j0; j < NHID / 4; j += 16) {
        v4f v = {};
        if (sidx >= 0) v = *(const v4f*)(src + j * 4);
        *(v4f*)(As + r * NHID + j * 4) = v;
      }
    }
    __syncthreads();

    for (int ntile = wave; ntile < 15; ntile += 8) {   // wave-uniform loop
      int n  = ntile * 16 + lane15;
      int nc = (n < NTPC) ? n : (NTPC - 1);            // clamp OOB lanes (masked at store)
      const float* arow = As + lane15 * NHID;
      v8f acc = {};
      for (int kb = 0; kb < NHID; kb += 4) {
        int k0 = kb + khalf;
        v2f a = *(const v2f*)(arow + k0);
        v2f b;
        b.x = Wc[(size_t)k0 * NTPC + nc];
        b.y = Wc[(size_t)(k0 + 1) * NTPC + nc];
        acc = __builtin_amdgcn_wmma_f32_16x16x4_f32(false, a, false, b,
                                                    (short)0, acc, false, false);
      }
      if (n < NTPC) {
        float bias = bb[(size_t)c * NTPC + n];
        int mhi = (lane >> 4) * 8;
#pragma unroll
        for (int r = 0; r < 8; ++r) {
          int m = r + mhi;
          if (base + m < cnt) {
            int sidx = list[c * BATCH + base + m];
            blogits[(size_t)sidx * NTPC + n] = acc[r] + bias;
          }
        }
      }
    }
    __syncthreads();
  }
}

// ---------------- bottom softmax + combine ----------------

__global__ void __launch_bounds__(256) hs_bottom_softmax(const float* __restrict__ blogits,
                                                         const int* __restrict__ labels,
                                                         const float* __restrict__ top_prob,
                                                         float* __restrict__ out) {
  __shared__ float sh[256];
  int s = blockIdx.x, t = threadIdx.x;
  float l = (t < NTPC) ? blogits[(size_t)s * NTPC + t] : -INFINITY;
  sh[t] = l; __syncthreads();
  for (int k = 128; k > 0; k >>= 1) { if (t < k) sh[t] = fmaxf(sh[t], sh[t + k]); __syncthreads(); }
  float mx = sh[0]; __syncthreads();
  float e = (t < NTPC) ? expf(l - mx) : 0.f;
  sh[t] = e; __syncthreads();
  for (int k = 128; k > 0; k >>= 1) { if (t < k) sh[t] += sh[t + k]; __syncthreads(); }
  float sum = sh[0];
  if (t == 0) {
    int kb = labels[s] % NTPC;
    out[s] = top_prob[s] * (expf(blogits[(size_t)s * NTPC + kb] - mx) / sum);
  }
}

// ---------------- launch ----------------

extern "C" void kernel_launch(void* const* d_in, const int* in_sizes, int n_in,
                              void* d_out, int out_size, void* d_ws, size_t ws_size,
                              hipStream_t stream) {
  const float* inputs = (const float*)d_in[0];
  const float* top_W  = (const float*)d_in[1];
  const float* top_b  = (const float*)d_in[2];
  const float* bot_W  = (const float*)d_in[3];
  const float* bot_b  = (const float*)d_in[4];
  const int*   labels = (const int*)d_in[5];
  float* out = (float*)d_out;

  // workspace layout
  float* ws_f       = (float*)d_ws;
  float* top_logits = ws_f;                                    // 1024*224
  float* top_prob   = top_logits + (size_t)BATCH * NCLS;       // 1024
  float* bot_logits = top_prob + BATCH;                        // 1024*225
  int*   counts     = (int*)(bot_logits + (size_t)BATCH * NTPC); // 224 (+pad)
  int*   list       = counts + 256;                            // 224*1024

  hipLaunchKernelGGL(hs_zero_counts, dim3(1), dim3(256), 0, stream, counts);
  hipLaunchKernelGGL(hs_bucket, dim3((BATCH + 255) / 256), dim3(256), 0, stream,
                     labels, counts, list);
  hipLaunchKernelGGL(hs_top_gemm, dim3((BATCH / 16) * (NCLS / 16)), dim3(32), 0, stream,
                     inputs, top_W, top_b, top_logits);
  hipLaunchKernelGGL(hs_top_softmax, dim3(BATCH), dim3(256), 0, stream,
                     top_logits, labels, top_prob);
  hipLaunchKernelGGL(hs_bottom_gemm, dim3(NCLS), dim3(256), 0, stream,
                     inputs, bot_W, bot_b, counts, list, bot_logits);
  hipLaunchKernelGGL(hs_bottom_softmax, dim3(BATCH), dim3(256), 0, stream,
                     bot_logits, labels, top_prob, out);
}